// NodeSampler_5669356830978
// MI455X (gfx1250) — compile-verified
//
#include <hip/hip_runtime.h>
#include <math.h>

// ---------------------------------------------------------------------------
// NodeSampler pipeline for MI455X (gfx1250, wave32, WMMA).
// Convs run as implicit GEMMs on V_WMMA_F32_16X16X4_F32 (full f32 precision,
// matches reference numerics). NMS / top-k / grid-sample are bandwidth-bound
// helper kernels.
// ---------------------------------------------------------------------------

#define B_    8
#define C_    256
#define H_    64
#define W_    64
#define HW_   4096
#define HN_   1024
#define NPIX_ 1048576     // 1024*1024
#define NN_   2048
#define NEG_INF (-__builtin_inff())

typedef float v2f __attribute__((ext_vector_type(2)));
typedef float v8f __attribute__((ext_vector_type(8)));

__device__ __forceinline__ v8f wmma_f32(v2f a, v2f b, v8f c) {
  // 8-arg pattern: (neg_a, A, neg_b, B, c_mod, C, reuse_a, reuse_b)
  return __builtin_amdgcn_wmma_f32_16x16x4_f32(false, a, false, b, 0, c, false, false);
}

// ---------------------------------------------------------------------------
// conv3x3 (pad=1) + exact GELU, NCHW.  Implicit GEMM: per batch
// M=4096 pixels, N=256 out-channels, K=256*9.  Block = 128 thr = 4 waves,
// block tile 64M x 64N, wave tile 16M x 64N (4 f32 accumulators).
// Weight tile (K=256 x N=64) for the current tap staged in 64KB LDS.
// Halo handling: addresses clamped into the image so loads are always legal
// and unconditional; padding zeros applied with cndmask on the value (avoids
// per-load EXEC save/restore in the hot loop).
// ---------------------------------------------------------------------------
__global__ __launch_bounds__(128) void k_conv3x3_gelu(
    const float* __restrict__ X, const float* __restrict__ W1,
    const float* __restrict__ B1, float* __restrict__ Y) {
  __shared__ float Bs[C_ * 64];
  const int lane = threadIdx.x & 31;
  const int wave = threadIdx.x >> 5;
  const int m0 = blockIdx.x * 64 + wave * 16;
  const int n0 = blockIdx.y * 64;
  const int b  = blockIdx.z;
  const int mrow = m0 + (lane & 15);
  const int h = mrow >> 6;          // constant across the 16-row tile
  const int w = mrow & 63;
  const int koff = (lane >> 4) << 1;  // lanes 16..31 hold K=2,3
  const float* Xb = X + (size_t)b * C_ * HW_;

  v8f acc[4];
  for (int j = 0; j < 4; ++j) {
    float bv = B1[n0 + j * 16 + (lane & 15)];
    for (int e = 0; e < 8; ++e) acc[j][e] = bv;
  }

  for (int tap = 0; tap < 9; ++tap) {
    const int dy = tap / 3 - 1, dx = tap % 3 - 1;
    __syncthreads();
    for (int t = threadIdx.x; t < C_ * 64; t += 128) {
      int k = t >> 6, nn = t & 63;
      Bs[t] = W1[((size_t)(n0 + nn) * C_ + k) * 9 + tap];
    }
    __syncthreads();
    const int hs = h + dy;
    const int wsrc = w + dx;
    const bool ok = (hs >= 0) && (hs < H_) && (wsrc >= 0) && (wsrc < W_);
    // clamped (always-valid) address; value zeroed by cndmask when !ok
    const int hsafe = min(max(hs, 0), H_ - 1);
    const int wsafe = min(max(wsrc, 0), W_ - 1);
    const float* Xp = Xb + hsafe * W_ + wsafe;
    for (int k0 = 0; k0 < C_; k0 += 4) {
      float v0 = Xp[(size_t)(k0 + koff) * HW_];
      float v1 = Xp[(size_t)(k0 + koff + 1) * HW_];
      v2f a;
      a[0] = ok ? v0 : 0.0f;
      a[1] = ok ? v1 : 0.0f;
#pragma unroll
      for (int j = 0; j < 4; ++j) {
        v2f bf;
        int nc = j * 16 + (lane & 15);
        bf[0] = Bs[(k0 + koff) * 64 + nc];
        bf[1] = Bs[(k0 + koff + 1) * 64 + nc];
        acc[j] = wmma_f32(a, bf, acc[j]);
      }
    }
  }
  // exact GELU epilogue, store NCHW
  for (int j = 0; j < 4; ++j) {
    int nc = n0 + j * 16 + (lane & 15);
    float* Yc = Y + ((size_t)b * C_ + nc) * HW_;
    for (int e = 0; e < 8; ++e) {
      int r = m0 + e + ((lane >> 4) << 3);
      float xv = acc[j][e];
      Yc[r] = 0.5f * xv * (1.0f + erff(xv * 0.70710678118654752440f));
    }
  }
}

// ---------------------------------------------------------------------------
// conv1x1 GEMM.  MODE 0: w2 path, epilogue does pixel-shuffle scatter into
// the (B,1024,1024) score map + border mask.  MODE 1: wf path, plain NCHW.
// ---------------------------------------------------------------------------
template <int MODE>
__global__ __launch_bounds__(128) void k_conv1x1(
    const float* __restrict__ X, const float* __restrict__ Wm,
    const float* __restrict__ Bv, float* __restrict__ Y) {
  __shared__ float Bs[C_ * 64];
  const int lane = threadIdx.x & 31;
  const int wave = threadIdx.x >> 5;
  const int m0 = blockIdx.x * 64 + wave * 16;
  const int n0 = blockIdx.y * 64;
  const int b  = blockIdx.z;
  const int mrow = m0 + (lane & 15);
  const int koff = (lane >> 4) << 1;
  const float* Xb = X + (size_t)b * C_ * HW_;

  for (int t = threadIdx.x; t < C_ * 64; t += 128) {
    int k = t >> 6, nn = t & 63;
    Bs[t] = Wm[(size_t)(n0 + nn) * C_ + k];
  }
  v8f acc[4];
  for (int j = 0; j < 4; ++j) {
    float bv = Bv[n0 + j * 16 + (lane & 15)];
    for (int e = 0; e < 8; ++e) acc[j][e] = bv;
  }
  __syncthreads();

  for (int k0 = 0; k0 < C_; k0 += 4) {
    v2f a;
    a[0] = Xb[(size_t)(k0 + koff) * HW_ + mrow];
    a[1] = Xb[(size_t)(k0 + koff + 1) * HW_ + mrow];
    if (k0 + 16 < C_)  // global_prefetch_b8 ahead in K
      __builtin_prefetch(&Xb[(size_t)(k0 + 16 + koff) * HW_ + mrow], 0, 1);
#pragma unroll
    for (int j = 0; j < 4; ++j) {
      v2f bf;
      int nc = j * 16 + (lane & 15);
      bf[0] = Bs[(k0 + koff) * 64 + nc];
      bf[1] = Bs[(k0 + koff + 1) * 64 + nc];
      acc[j] = wmma_f32(a, bf, acc[j]);
    }
  }

  for (int j = 0; j < 4; ++j) {
    int nc = n0 + j * 16 + (lane & 15);
    for (int e = 0; e < 8; ++e) {
      int r = m0 + e + ((lane >> 4) << 3);
      float v = acc[j][e];
      if (MODE == 0) {
        // pixel shuffle: channel c = sy*16+sx ; Y = h*16+sy, X = w*16+sx
        int hh = r >> 6, ww = r & 63;
        int sy = nc >> 4, sx = nc & 15;
        int Yr = hh * 16 + sy, Xc = ww * 16 + sx;
        if (Yr < 4 || Yr >= HN_ - 4 || Xc < 4 || Xc >= HN_ - 4) v = NEG_INF;
        Y[(size_t)b * NPIX_ + (size_t)Yr * HN_ + Xc] = v;
      } else {
        Y[((size_t)b * C_ + nc) * HW_ + r] = v;
      }
    }
  }
}

// ---------------------------------------------------------------------------
// Separable 9-wide max (f32) and or (u8) pools; -inf / 0 padding.
// ---------------------------------------------------------------------------
__global__ __launch_bounds__(256) void k_rowmax9(const float* __restrict__ in,
                                                 float* __restrict__ out) {
  int i = blockIdx.x * 256 + threadIdx.x;
  if (i >= B_ * NPIX_) return;
  int x = i & (HN_ - 1);
  const float* row = in + (i - x);
  float m = NEG_INF;
#pragma unroll
  for (int d = -4; d <= 4; ++d) {
    int xx = x + d;
    if (xx >= 0 && xx < HN_) m = fmaxf(m, row[xx]);
  }
  out[i] = m;
}

__global__ __launch_bounds__(256) void k_colmax9(const float* __restrict__ in,
                                                 float* __restrict__ out) {
  int i = blockIdx.x * 256 + threadIdx.x;
  if (i >= B_ * NPIX_) return;
  int x = i & (HN_ - 1);
  int y = (i >> 10) & (HN_ - 1);
  size_t base = (size_t)(i >> 20) * NPIX_;
  float m = NEG_INF;
#pragma unroll
  for (int d = -4; d <= 4; ++d) {
    int yy = y + d;
    if (yy >= 0 && yy < HN_) m = fmaxf(m, in[base + (size_t)yy * HN_ + x]);
  }
  out[i] = m;
}

__global__ __launch_bounds__(256) void k_rowor9(const unsigned char* __restrict__ in,
                                                unsigned char* __restrict__ out) {
  int i = blockIdx.x * 256 + threadIdx.x;
  if (i >= B_ * NPIX_) return;
  int x = i & (HN_ - 1);
  const unsigned char* row = in + (i - x);
  unsigned int m = 0;
#pragma unroll
  for (int d = -4; d <= 4; ++d) {
    int xx = x + d;
    if (xx >= 0 && xx < HN_) m |= row[xx];
  }
  out[i] = (unsigned char)(m != 0);
}

__global__ __launch_bounds__(256) void k_color9(const unsigned char* __restrict__ in,
                                                unsigned char* __restrict__ out) {
  int i = blockIdx.x * 256 + threadIdx.x;
  if (i >= B_ * NPIX_) return;
  int x = i & (HN_ - 1);
  int y = (i >> 10) & (HN_ - 1);
  size_t base = (size_t)(i >> 20) * NPIX_;
  unsigned int m = 0;
#pragma unroll
  for (int d = -4; d <= 4; ++d) {
    int yy = y + d;
    if (yy >= 0 && yy < HN_) m |= in[base + (size_t)yy * HN_ + x];
  }
  out[i] = (unsigned char)(m != 0);
}

// ---------------------------------------------------------------------------
// simple_nms elementwise stages
// ---------------------------------------------------------------------------
__global__ __launch_bounds__(256) void k_eqmask(const float* __restrict__ S,
                                                const float* __restrict__ T,
                                                unsigned char* __restrict__ M) {
  int i = blockIdx.x * 256 + threadIdx.x;
  if (i >= B_ * NPIX_) return;
  M[i] = (S[i] == T[i]) ? 1 : 0;
}

__global__ __launch_bounds__(256) void k_suppscores(const float* __restrict__ S,
                                                    const unsigned char* __restrict__ SUPP,
                                                    float* __restrict__ SS) {
  int i = blockIdx.x * 256 + threadIdx.x;
  if (i >= B_ * NPIX_) return;
  SS[i] = SUPP[i] ? NEG_INF : S[i];
}

__global__ __launch_bounds__(256) void k_update(const float* __restrict__ SS,
                                                const float* __restrict__ T,
                                                const unsigned char* __restrict__ SUPP,
                                                unsigned char* __restrict__ M) {
  int i = blockIdx.x * 256 + threadIdx.x;
  if (i >= B_ * NPIX_) return;
  unsigned char nm = ((SS[i] == T[i]) && !SUPP[i]) ? 1 : 0;
  M[i] = (unsigned char)(M[i] | nm);
}

__global__ __launch_bounds__(256) void k_finalize(const float* __restrict__ S,
                                                  const unsigned char* __restrict__ M,
                                                  float* __restrict__ F) {
  int i = blockIdx.x * 256 + threadIdx.x;
  if (i >= B_ * NPIX_) return;
  F[i] = M[i] ? S[i] : NEG_INF;
}

// ---------------------------------------------------------------------------
// Per-batch top-2048 of 1M: 4-pass radix select on monotone u32 keys,
// LDS compaction, then LDS bitonic sort (descending value, ascending index).
// ---------------------------------------------------------------------------
__global__ __launch_bounds__(1024) void k_topk(const float* __restrict__ F,
                                               float* __restrict__ w_out,
                                               float* __restrict__ nxy_out,
                                               int* __restrict__ sel) {
  const int b = blockIdx.x;
  const float* Fb = F + (size_t)b * NPIX_;
  __shared__ unsigned int hist[256];
  __shared__ unsigned long long pairs[NN_];
  __shared__ unsigned int s_prefix, s_krem, s_cnt1, s_cnt2;
  const int tid = threadIdx.x;

  if (tid == 0) { s_prefix = 0u; s_krem = NN_; }
  for (int pass = 0; pass < 4; ++pass) {
    const int shift = 24 - pass * 8;
    if (tid < 256) hist[tid] = 0u;
    __syncthreads();
    const unsigned int pref = s_prefix;
    for (int i = tid; i < NPIX_; i += 1024) {
      unsigned int u = __float_as_uint(Fb[i]);
      unsigned int key = (u & 0x80000000u) ? ~u : (u ^ 0x80000000u);
      if (pass == 0 || (key >> (shift + 8)) == (pref >> (shift + 8)))
        atomicAdd(&hist[(key >> shift) & 255u], 1u);
    }
    __syncthreads();
    if (tid == 0) {
      unsigned int krem = s_krem, cum = 0u;
      int bin = 255;
      for (; bin > 0; --bin) {
        unsigned int c = hist[bin];
        if (cum + c >= krem) break;
        cum += c;
      }
      s_krem = krem - cum;
      s_prefix = pref | ((unsigned int)bin << shift);
    }
    __syncthreads();
  }
  const unsigned int Tkey = s_prefix;
  const unsigned int ties = s_krem;          // # elements equal to Tkey to take
  const unsigned int cntG = NN_ - ties;      // # elements strictly greater
  if (tid == 0) { s_cnt1 = 0u; s_cnt2 = 0u; }
  __syncthreads();
  for (int i = tid; i < NPIX_; i += 1024) {
    unsigned int u = __float_as_uint(Fb[i]);
    unsigned int key = (u & 0x80000000u) ? ~u : (u ^ 0x80000000u);
    unsigned long long pk =
        ((unsigned long long)key << 32) | (unsigned int)(0xFFFFFFFFu - (unsigned int)i);
    if (key > Tkey) {
      unsigned int p = atomicAdd(&s_cnt1, 1u);
      pairs[p] = pk;
    } else if (key == Tkey) {
      unsigned int t = atomicAdd(&s_cnt2, 1u);
      if (t < ties) pairs[cntG + t] = pk;
    }
  }
  __syncthreads();
  // bitonic sort, descending (key desc, index asc via ~idx in low bits)
  for (unsigned int k = 2; k <= NN_; k <<= 1) {
    for (unsigned int j = k >> 1; j > 0; j >>= 1) {
      for (unsigned int i = tid; i < NN_; i += 1024) {
        unsigned int ixj = i ^ j;
        if (ixj > i) {
          unsigned long long a = pairs[i], c = pairs[ixj];
          bool up = ((i & k) == 0u);
          if (up ? (a < c) : (a > c)) { pairs[i] = c; pairs[ixj] = a; }
        }
      }
      __syncthreads();
    }
  }
  for (unsigned int jo = tid; jo < NN_; jo += 1024) {
    unsigned long long pk = pairs[jo];
    unsigned int key = (unsigned int)(pk >> 32);
    unsigned int idx = 0xFFFFFFFFu - (unsigned int)(pk & 0xFFFFFFFFull);
    unsigned int u = (key & 0x80000000u) ? (key ^ 0x80000000u) : ~key;
    w_out[(size_t)b * NN_ + jo] = __uint_as_float(u);
    int yy = (int)(idx >> 10), xx = (int)(idx & (HN_ - 1));
    float nx = (float)xx / (float)(HN_ - 1) * 2.0f - 1.0f;
    float ny = (float)yy / (float)(HN_ - 1) * 2.0f - 1.0f;
    size_t o = ((size_t)b * NN_ + jo) * 2;
    nxy_out[o] = nx;
    nxy_out[o + 1] = ny;
    sel[(size_t)b * NN_ + jo] = (int)idx;
  }
}

// ---------------------------------------------------------------------------
// grid_sample bilinear, align_corners=False, zeros padding.
// block = 256 threads (one per channel), grid = (2048 nodes, 8 batches).
// ---------------------------------------------------------------------------
__global__ __launch_bounds__(256) void k_gridsample(const float* __restrict__ FEAT,
                                                    const int* __restrict__ sel,
                                                    float* __restrict__ f_out) {
  const int n = blockIdx.x, b = blockIdx.y, c = threadIdx.x;
  const int idx = sel[(size_t)b * NN_ + n];
  const int yy = idx >> 10, xx = idx & (HN_ - 1);
  const float gx = (float)xx / (float)(HN_ - 1) * 2.0f - 1.0f;
  const float gy = (float)yy / (float)(HN_ - 1) * 2.0f - 1.0f;
  const float ix = ((gx + 1.0f) * (float)W_ - 1.0f) * 0.5f;
  const float iy = ((gy + 1.0f) * (float)H_ - 1.0f) * 0.5f;
  const float fx0 = floorf(ix), fy0 = floorf(iy);
  const int x0 = (int)fx0, y0 = (int)fy0;
  const int x1 = x0 + 1, y1 = y0 + 1;
  const float wx1 = ix - fx0, wx0 = 1.0f - wx1;
  const float wy1 = iy - fy0, wy0 = 1.0f - wy1;
  const float* fb = FEAT + ((size_t)b * C_ + c) * HW_;
  float acc = 0.0f;
#define TAP(xc, yc, wt)                                              \
  do {                                                               \
    if ((xc) >= 0 && (xc) < W_ && (yc) >= 0 && (yc) < H_)            \
      acc += fb[(yc) * W_ + (xc)] * (wt);                            \
  } while (0)
  TAP(x0, y0, wx0 * wy0);
  TAP(x1, y0, wx1 * wy0);
  TAP(x0, y1, wx0 * wy1);
  TAP(x1, y1, wx1 * wy1);
#undef TAP
  f_out[((size_t)b * NN_ + n) * C_ + c] = acc;
}

// ---------------------------------------------------------------------------
extern "C" void kernel_launch(void* const* d_in, const int* in_sizes, int n_in,
                              void* d_out, int out_size, void* d_ws, size_t ws_size,
                              hipStream_t stream) {
  (void)in_sizes; (void)n_in; (void)out_size; (void)ws_size;
  const float* featmap = (const float*)d_in[0];
  const float* w1 = (const float*)d_in[1];
  const float* b1 = (const float*)d_in[2];
  const float* w2 = (const float*)d_in[3];
  const float* b2 = (const float*)d_in[4];
  const float* wf = (const float*)d_in[5];
  const float* bf = (const float*)d_in[6];

  char* ws = (char*)d_ws;
  const size_t MB32 = 33554432ull;               // 8*1024*1024 floats
  float* WM1 = (float*)(ws + 0 * MB32);          // conv1 out; reused as SS
  float* S   = (float*)(ws + 1 * MB32);          // score map
  float* R   = (float*)(ws + 2 * MB32);          // row-pool tmp; reused as FEAT
  float* T   = (float*)(ws + 3 * MB32);          // col-pool out; reused as final F
  unsigned char* M    = (unsigned char*)(ws + 4 * MB32);
  unsigned char* SUPP = M + (size_t)B_ * NPIX_;
  unsigned char* U8A  = SUPP + (size_t)B_ * NPIX_;
  int* SEL = (int*)(U8A + (size_t)B_ * NPIX_);
  float* SS = WM1;
  float* FEAT = R;
  float* F = T;

  float* out_nxy = (float*)d_out;                      // (B, 2048, 2)
  float* out_w   = out_nxy + (size_t)B_ * NN_ * 2;     // (B, 2048)
  float* out_f   = out_w + (size_t)B_ * NN_;           // (B, 2048, 256)

  dim3 gG(HW_ / 64, C_ / 64, B_);   // (64, 4, 8)
  const int NE = B_ * NPIX_;
  const int gE = NE / 256;

  // stage 1: conv3x3 + GELU
  k_conv3x3_gelu<<<gG, 128, 0, stream>>>(featmap, w1, b1, WM1);
  // stage 2: conv1x1 (w2) fused pixel-shuffle + border mask -> scores
  k_conv1x1<0><<<gG, 128, 0, stream>>>(WM1, w2, b2, S);

  // stage 3: simple_nms (r=4), exact reference semantics
  k_rowmax9<<<gE, 256, 0, stream>>>(S, R);
  k_colmax9<<<gE, 256, 0, stream>>>(R, T);
  k_eqmask <<<gE, 256, 0, stream>>>(S, T, M);
  for (int it = 0; it < 2; ++it) {
    k_rowor9    <<<gE, 256, 0, stream>>>(M, U8A);
    k_color9    <<<gE, 256, 0, stream>>>(U8A, SUPP);
    k_suppscores<<<gE, 256, 0, stream>>>(S, SUPP, SS);
    k_rowmax9   <<<gE, 256, 0, stream>>>(SS, R);
    k_colmax9   <<<gE, 256, 0, stream>>>(R, T);
    k_update    <<<gE, 256, 0, stream>>>(SS, T, SUPP, M);
  }
  k_finalize<<<gE, 256, 0, stream>>>(S, M, F);

  // stage 4: conv1x1 (wf) -> feature map (R is free after NMS)
  k_conv1x1<1><<<gG, 128, 0, stream>>>(featmap, wf, bf, FEAT);

  // stage 5: per-batch top-2048 (+ norm coords) and bilinear gather
  k_topk<<<B_, 1024, 0, stream>>>(F, out_w, out_nxy, SEL);
  k_gridsample<<<dim3(NN_, B_), 256, 0, stream>>>(FEAT, SEL, out_f);
}